// SpatialTemporalGraphTransformer_63694365000322
// MI455X (gfx1250) — compile-verified
//
#include <hip/hip_runtime.h>
#include <hip/hip_bf16.h>
#include <math.h>

// ---------------------------------------------------------------------------
// SpatialTemporalGraphTransformer forward for MI455X (gfx1250, wave32, WMMA)
// All GEMMs use V_WMMA_F32_16X16X4_F32 (fp32 in/out, matches reference dtype).
// dec2 (403MB weight stream @ 23.3TB/s ~ 17us) is the bandwidth-dominant op:
// one wave owns a full 48-row x 16-col strip so each weight row is streamed
// exactly once (3 WMMAs share one B fragment), with NT prefetch ahead.
// ---------------------------------------------------------------------------

typedef float v2f __attribute__((ext_vector_type(2)));
typedef float v8f __attribute__((ext_vector_type(8)));

static constexpr int NN    = 3200;      // nodes = B*T*V
static constexpr int EE    = 102400;    // edges
static constexpr int EP    = EE + NN;   // edges + self loops
static constexpr int DM    = 256;       // model dim
static constexpr int DIN   = 64;        // input feature dim
static constexpr int BB    = 2;
static constexpr int LH    = 1600;      // T*V
static constexpr int DFF   = 2048;
static constexpr int TOUT  = 24;        // T*UPSAMPLE
static constexpr int M2    = BB * TOUT; // 48 decoder rows
static constexpr int NOUT  = 256 * 256 * 3; // 196608
static constexpr float LN_EPS = 1e-5f;

#define ACT_NONE 0
#define ACT_RELU 1

// ---------------- small helpers -------------------------------------------

__device__ __forceinline__ void atomAddF(float* p, float v) {
  unsafeAtomicAdd(p, v);   // global_atomic_add_f32
}

__device__ __forceinline__ void atomicMaxF(float* addr, float val) {
  unsigned int* ua = (unsigned int*)addr;
  unsigned int old = *ua;
  while (__uint_as_float(old) < val) {
    unsigned int assumed = old;
    old = atomicCAS(ua, assumed, __float_as_uint(val));
    if (old == assumed) break;
  }
}

__device__ __forceinline__ float waveSum(float v) {
#pragma unroll
  for (int o = 16; o > 0; o >>= 1) v += __shfl_xor(v, o, 32);
  return v;
}

__global__ void fill_f32(float* p, float v, int n) {
  int i = blockIdx.x * blockDim.x + threadIdx.x;
  if (i < n) p[i] = v;
}

// ---------------- generic fp32 WMMA GEMM ----------------------------------
// C[M,N] = act(A[M,K] @ W[N,K]^T + bias).  One wave per 16x16 tile.
// A-frag (16x4): lane%16 = M row, lane/16 selects K pair {k,k+1}/{k+2,k+3}.
// B-frag (4x16): lane%16 = N col, same K-pair split (mirror of A layout).
__global__ void gemm_wmma_f32(const float* __restrict__ A,
                              const float* __restrict__ Bw,
                              const float* __restrict__ bias,
                              float* __restrict__ C,
                              int M, int K, int Nn, int act) {
  int wid  = threadIdx.x >> 5;
  int lane = threadIdx.x & 31;
  int ntn  = Nn >> 4;
  int ntiles = (M >> 4) * ntn;
  int tile = blockIdx.x * (blockDim.x >> 5) + wid;
  if (tile >= ntiles) return;                    // uniform per wave: EXEC stays all-1
  int tm = tile / ntn, tn = tile % ntn;
  int m0 = tm << 4, n0 = tn << 4;
  int half = lane >> 4;
  int l16  = lane & 15;
  const float* arow = A  + (size_t)(m0 + l16) * K + (half << 1);
  const float* brow = Bw + (size_t)(n0 + l16) * K + (half << 1);
  v8f c = {0.f,0.f,0.f,0.f,0.f,0.f,0.f,0.f};
  for (int k = 0; k < K; k += 4) {
    __builtin_prefetch(brow + k + 64, 0, 3);     // weights reused across M tiles: keep near
    v2f a = *(const v2f*)(arow + k);
    v2f b = *(const v2f*)(brow + k);
    c = __builtin_amdgcn_wmma_f32_16x16x4_f32(false, a, false, b, (short)0, c,
                                              false, false);
  }
  int col = n0 + l16;
  float bs = bias ? bias[col] : 0.f;
#pragma unroll
  for (int i = 0; i < 8; i++) {
    int row = m0 + half * 8 + i;                 // C layout: VGPR i -> M = i (+8 for hi half)
    float v = c[i] + bs;
    if (act == ACT_RELU) v = fmaxf(v, 0.f);
    C[(size_t)row * Nn + col] = v;
  }
}

// ---------------- dec2 GEMM + fused transpose store -----------------------
// out[b,h,w,c,t] = (v1[48,512] @ dec2_w[196608,512]^T + dec2_b)[b*24+t, (h*256+w)*3+c]
// One wave per 16-column tile computes the whole M=48 strip: the 403MB weight
// matrix is streamed exactly once; each B fragment feeds 3 WMMAs.
__global__ void gemm_dec2_wmma(const float* __restrict__ A,
                               const float* __restrict__ Bw,
                               const float* __restrict__ bias,
                               float* __restrict__ out) {
  const int K = 512, ntn = NOUT >> 4;            // 12288 column tiles
  int wid = threadIdx.x >> 5, lane = threadIdx.x & 31;
  int tn = blockIdx.x * (blockDim.x >> 5) + wid;
  if (tn >= ntn) return;
  int n0 = tn << 4;
  int half = lane >> 4, l16 = lane & 15;
  const float* a0 = A + (size_t)(0  + l16) * K + (half << 1);
  const float* a1 = A + (size_t)(16 + l16) * K + (half << 1);
  const float* a2 = A + (size_t)(32 + l16) * K + (half << 1);
  const float* brow = Bw + (size_t)(n0 + l16) * K + (half << 1);
  v8f c0 = {0.f,0.f,0.f,0.f,0.f,0.f,0.f,0.f};
  v8f c1 = c0, c2 = c0;
  for (int k = 0; k < K; k += 4) {
    __builtin_prefetch(brow + k + 128, 0, 0);    // stream-once weights: non-temporal
    v2f b  = *(const v2f*)(brow + k);
    v2f f0 = *(const v2f*)(a0 + k);
    v2f f1 = *(const v2f*)(a1 + k);
    v2f f2 = *(const v2f*)(a2 + k);
    c0 = __builtin_amdgcn_wmma_f32_16x16x4_f32(false, f0, false, b, (short)0, c0,
                                               false, false);
    c1 = __builtin_amdgcn_wmma_f32_16x16x4_f32(false, f1, false, b, (short)0, c1,
                                               false, false);
    c2 = __builtin_amdgcn_wmma_f32_16x16x4_f32(false, f2, false, b, (short)0, c2,
                                               false, false);
  }
  int n = n0 + l16;
  float bs = bias[n];
  int hh  = n / 768;
  int rem = n - hh * 768;
  int ww  = rem / 3;
  int cc  = rem - ww * 3;
  size_t obase = ((((size_t)0 * 256 + hh) * 256 + ww) * 3 + cc) * TOUT;
  size_t ostep = (size_t)256 * 256 * 3 * TOUT;   // +1 in batch dim
#pragma unroll
  for (int j = 0; j < 3; j++) {
    v8f& c = (j == 0) ? c0 : (j == 1) ? c1 : c2;
#pragma unroll
    for (int i = 0; i < 8; i++) {
      int m = j * 16 + half * 8 + i;
      int b = m / TOUT, t = m - b * TOUT;
      out[obase + (size_t)b * ostep + t] = c[i] + bs;
    }
  }
}

// ---------------- GATv2 kernels -------------------------------------------

__global__ void gat_selfloop_accum(const int* __restrict__ ei,
                                   const float* __restrict__ ea,
                                   float* cnt, float* la) {
  int e = blockIdx.x * blockDim.x + threadIdx.x;
  if (e >= EE) return;
  int d = ei[EE + e];
  atomAddF(cnt + d, 1.f);
  atomAddF(la + 2 * d,     ea[2 * e]);
  atomAddF(la + 2 * d + 1, ea[2 * e + 1]);
}

__global__ void gat_selfloop_div(float* cnt, float* la) {
  int n = blockIdx.x * blockDim.x + threadIdx.x;
  if (n >= NN) return;
  float c = fmaxf(cnt[n], 1.f);
  la[2 * n]     /= c;
  la[2 * n + 1] /= c;
}

// wave per edge: logit[e] = att . leaky_relu(xl[src]+xr[dst]+ea@We^T)
__global__ void gat_logit(const int* __restrict__ ei, const float* __restrict__ ea,
                          const float* __restrict__ la,
                          const float* __restrict__ xl, const float* __restrict__ xr,
                          const float* __restrict__ We, const float* __restrict__ att,
                          float* __restrict__ logit, float* __restrict__ mm) {
  int e = blockIdx.x * (blockDim.x >> 5) + (threadIdx.x >> 5);
  int lane = threadIdx.x & 31;
  if (e >= EP) return;
  int s, d; float a0, a1;
  if (e < EE) { s = ei[e]; d = ei[EE + e]; a0 = ea[2*e]; a1 = ea[2*e+1]; }
  else        { s = d = e - EE; a0 = la[2*(e-EE)]; a1 = la[2*(e-EE)+1]; }
  const float* xls = xl + (size_t)s * DM;
  const float* xrd = xr + (size_t)d * DM;
  float acc = 0.f;
#pragma unroll
  for (int i = 0; i < 8; i++) {
    int dd = lane + 32 * i;
    float t = xls[dd] + xrd[dd] + a0 * We[2*dd] + a1 * We[2*dd+1];
    t = (t > 0.f) ? t : 0.2f * t;
    acc += t * att[dd];
  }
  acc = waveSum(acc);
  if (lane == 0) { logit[e] = acc; atomicMaxF(mm + d, acc); }
}

__global__ void gat_wexp(const int* __restrict__ ei, const float* __restrict__ logit,
                         const float* __restrict__ mm, float* __restrict__ wex,
                         float* __restrict__ den) {
  int e = blockIdx.x * blockDim.x + threadIdx.x;
  if (e >= EP) return;
  int d = (e < EE) ? ei[EE + e] : (e - EE);
  float w = __expf(logit[e] - mm[d]);
  wex[e] = w;
  atomAddF(den + d, w);
}

// wave per edge: out[dst] += alpha * xl[src]
__global__ void gat_scatter(const int* __restrict__ ei, const float* __restrict__ wex,
                            const float* __restrict__ den, const float* __restrict__ xl,
                            float* __restrict__ out) {
  int e = blockIdx.x * (blockDim.x >> 5) + (threadIdx.x >> 5);
  int lane = threadIdx.x & 31;
  if (e >= EP) return;
  int s, d;
  if (e < EE) { s = ei[e]; d = ei[EE + e]; } else { s = d = e - EE; }
  float alpha = wex[e] / den[d];
  const float* xls = xl + (size_t)s * DM;
  float* od = out + (size_t)d * DM;
#pragma unroll
  for (int i = 0; i < 8; i++) {
    int dd = lane + 32 * i;
    atomAddF(od + dd, alpha * xls[dd]);
  }
}

__global__ void gat_bias_silu(float* h, const float* __restrict__ b, int n, int do_silu) {
  int i = blockIdx.x * blockDim.x + threadIdx.x;
  if (i >= n) return;
  float v = h[i] + b[i & (DM - 1)];
  if (do_silu) v = v / (1.f + __expf(-v));
  h[i] = v;
}

// ---------------- transformer encoder kernels -----------------------------

// flash-style attention, one thread per (b, head, query), dh=32 in registers
__global__ void attn_kernel(const float* __restrict__ qkv, float* __restrict__ out,
                            int L) {
  int l = blockIdx.x * blockDim.x + threadIdx.x;
  int h = blockIdx.y, b = blockIdx.z;
  if (l >= L) return;
  const float scale = 0.17677669529663687f; // 1/sqrt(32)
  const float* qr = qkv + ((size_t)(b * L + l)) * 768 + h * 32;
  float qv[32];
#pragma unroll
  for (int i = 0; i < 32; i++) qv[i] = qr[i] * scale;
  float Mx = -3.4e38f, S = 0.f, acc[32];
#pragma unroll
  for (int i = 0; i < 32; i++) acc[i] = 0.f;
  for (int m = 0; m < L; m++) {
    const float* kr = qkv + ((size_t)(b * L + m)) * 768 + 256 + h * 32;
    const float* vr = kr + 256;
    float s = 0.f;
#pragma unroll
    for (int i = 0; i < 32; i++) s += qv[i] * kr[i];
    float nM = fmaxf(Mx, s);
    float corr = __expf(Mx - nM);
    float p = __expf(s - nM);
    S = S * corr + p;
#pragma unroll
    for (int i = 0; i < 32; i++) acc[i] = acc[i] * corr + p * vr[i];
    Mx = nM;
  }
  float inv = 1.f / S;
  float* orow = out + ((size_t)(b * L + l)) * 256 + h * 32;
#pragma unroll
  for (int i = 0; i < 32; i++) orow[i] = acc[i] * inv;
}

// out = LN(x + r) ; wave per row, in-place safe (row cached in regs)
__global__ void add_ln(const float* __restrict__ x, const float* __restrict__ r,
                       const float* __restrict__ g, const float* __restrict__ bt,
                       float* __restrict__ out, int rows) {
  int row = blockIdx.x * (blockDim.x >> 5) + (threadIdx.x >> 5);
  int lane = threadIdx.x & 31;
  if (row >= rows) return;
  const float* xr = x + (size_t)row * DM;
  const float* rr = r + (size_t)row * DM;
  float v[8], s = 0.f;
#pragma unroll
  for (int i = 0; i < 8; i++) { v[i] = xr[lane + 32*i] + rr[lane + 32*i]; s += v[i]; }
  s = waveSum(s);
  float mean = s * (1.f / DM);
  float var = 0.f;
#pragma unroll
  for (int i = 0; i < 8; i++) { float d = v[i] - mean; var += d * d; }
  var = waveSum(var) * (1.f / DM);
  float inv = rsqrtf(var + LN_EPS);
  float* orow = out + (size_t)row * DM;
#pragma unroll
  for (int i = 0; i < 8; i++) {
    int dd = lane + 32 * i;
    orow[dd] = (v[i] - mean) * inv * g[dd] + bt[dd];
  }
}

__global__ void ctx_mean(const float* __restrict__ h, float* __restrict__ ctx, int L) {
  int t = blockIdx.x * blockDim.x + threadIdx.x;
  if (t >= BB * DM) return;
  int b = t >> 8, d = t & 255;
  float s = 0.f;
  for (int l = 0; l < L; l++) s += h[((size_t)(b * L + l)) * DM + d];
  ctx[t] = s / (float)L;
}

__global__ void q_init(const float* __restrict__ queries, const float* __restrict__ ctx,
                       float* __restrict__ q) {
  int i = blockIdx.x * blockDim.x + threadIdx.x;
  if (i >= M2 * DM) return;
  int row = i >> 8, d = i & 255;
  int b = row / TOUT, t = row - b * TOUT;
  q[i] = queries[t * DM + d] + 0.01f * ctx[b * DM + d];
}

// ---------------- host orchestration --------------------------------------

struct EncP { const float *ff1_b,*ff1_w,*ff2_b,*ff2_w,*in_b,*in_w,
                          *ln1_b,*ln1_g,*ln2_b,*ln2_g,*out_b,*out_w; };
struct GatP { const float *We,*Wl,*Wr,*att,*b,*bl,*br; };

static inline void launch_gemm(const float* A, const float* W, const float* bias,
                               float* C, int M, int K, int Nn, int act,
                               hipStream_t s) {
  int tiles = (M / 16) * (Nn / 16);
  int blocks = (tiles + 7) / 8;
  gemm_wmma_f32<<<blocks, 256, 0, s>>>(A, W, bias, C, M, K, Nn, act);
}

extern "C" void kernel_launch(void* const* d_in, const int* in_sizes, int n_in,
                              void* d_out, int out_size, void* d_ws, size_t ws_size,
                              hipStream_t stream) {
  // ---- inputs (setup_inputs insertion order; params as jax tree leaves,
  //      dicts alphabetical, lists in order) ----
  const float* x  = (const float*)d_in[0];
  const int*   ei = (const int*)d_in[1];
  const float* ea = (const float*)d_in[2];
  int ix = 3;
  auto F = [&](void) { return (const float*)d_in[ix++]; };
  const float* dec1_b = F(); const float* dec1_w = F();
  const float* dec2_b = F(); const float* dec2_w = F();
  EncP enc[2];
  for (int e = 0; e < 2; e++) {
    enc[e].ff1_b = F(); enc[e].ff1_w = F(); enc[e].ff2_b = F(); enc[e].ff2_w = F();
    enc[e].in_b  = F(); enc[e].in_w  = F();
    enc[e].ln1_b = F(); enc[e].ln1_g = F(); enc[e].ln2_b = F(); enc[e].ln2_g = F();
    enc[e].out_b = F(); enc[e].out_w = F();
  }
  GatP gat[2];
  for (int g = 0; g < 2; g++) {
    gat[g].We = F(); gat[g].Wl = F(); gat[g].Wr = F(); gat[g].att = F();
    gat[g].b  = F(); gat[g].bl = F(); gat[g].br = F();
  }
  const float* queries = F();
  float* out = (float*)d_out;

  // ---- scratch carve (floats) ----
  float* W = (float*)d_ws;
  size_t o = 0;
  float* xl   = W + o; o += (size_t)NN * DM;
  float* xr   = W + o; o += (size_t)NN * DM;
  float* hA   = W + o; o += (size_t)NN * DM;   // gat1 out (silu)
  float* hB   = W + o; o += (size_t)NN * DM;   // gat2 out -> encoder state
  float* qkv  = W + o; o += (size_t)NN * 768;
  float* attn = W + o; o += (size_t)NN * DM;
  float* proj = W + o; o += (size_t)NN * DM;
  float* ffm  = W + o; o += (size_t)NN * DFF;
  float* cnt  = W + o; o += NN;
  float* la   = W + o; o += 2 * NN;
  float* logit= W + o; o += EP;
  float* wex  = W + o; o += EP;
  float* mm   = W + o; o += NN;
  float* den  = W + o; o += NN;
  float* ctx  = W + o; o += BB * DM;
  float* qb   = W + o; o += (size_t)M2 * DM;
  float* v1   = W + o; o += (size_t)M2 * 512;

  auto zero = [&](float* p, size_t n) {
    fill_f32<<<(unsigned)((n + 255) / 256), 256, 0, stream>>>(p, 0.f, (int)n);
  };

  // ---- GAT self-loop mean edge_attr (shared by both layers) ----
  zero(cnt, NN); zero(la, 2 * NN);
  gat_selfloop_accum<<<(EE + 255) / 256, 256, 0, stream>>>(ei, ea, cnt, la);
  gat_selfloop_div<<<(NN + 255) / 256, 256, 0, stream>>>(cnt, la);

  const int EPB = (EP + 7) / 8;   // wave-per-edge blocks (8 waves / block)
  // ---- GAT layer 1 (in 64 -> 256) ----
  launch_gemm(x, gat[0].Wl, gat[0].bl, xl, NN, DIN, DM, ACT_NONE, stream);
  launch_gemm(x, gat[0].Wr, gat[0].br, xr, NN, DIN, DM, ACT_NONE, stream);
  fill_f32<<<(NN + 255) / 256, 256, 0, stream>>>(mm, -3.4e38f, NN);
  zero(den, NN); zero(hA, (size_t)NN * DM);
  gat_logit<<<EPB, 256, 0, stream>>>(ei, ea, la, xl, xr, gat[0].We, gat[0].att, logit, mm);
  gat_wexp<<<(EP + 255) / 256, 256, 0, stream>>>(ei, logit, mm, wex, den);
  gat_scatter<<<EPB, 256, 0, stream>>>(ei, wex, den, xl, hA);
  gat_bias_silu<<<(NN * DM + 255) / 256, 256, 0, stream>>>(hA, gat[0].b, NN * DM, 1);

  // ---- GAT layer 2 (256 -> 256) ----
  launch_gemm(hA, gat[1].Wl, gat[1].bl, xl, NN, DM, DM, ACT_NONE, stream);
  launch_gemm(hA, gat[1].Wr, gat[1].br, xr, NN, DM, DM, ACT_NONE, stream);
  fill_f32<<<(NN + 255) / 256, 256, 0, stream>>>(mm, -3.4e38f, NN);
  zero(den, NN); zero(hB, (size_t)NN * DM);
  gat_logit<<<EPB, 256, 0, stream>>>(ei, ea, la, xl, xr, gat[1].We, gat[1].att, logit, mm);
  gat_wexp<<<(EP + 255) / 256, 256, 0, stream>>>(ei, logit, mm, wex, den);
  gat_scatter<<<EPB, 256, 0, stream>>>(ei, wex, den, xl, hB);
  gat_bias_silu<<<(NN * DM + 255) / 256, 256, 0, stream>>>(hB, gat[1].b, NN * DM, 0);

  // ---- transformer encoder layer (post-LN, ReLU FFN) ----
  auto enc_layer = [&](float* xbuf, int rows, int L, const EncP& P) {
    launch_gemm(xbuf, P.in_w, P.in_b, qkv, rows, DM, 768, ACT_NONE, stream);
    attn_kernel<<<dim3((L + 127) / 128, 8, BB), 128, 0, stream>>>(qkv, attn, L);
    launch_gemm(attn, P.out_w, P.out_b, proj, rows, DM, DM, ACT_NONE, stream);
    add_ln<<<(rows + 7) / 8, 256, 0, stream>>>(xbuf, proj, P.ln1_g, P.ln1_b, xbuf, rows);
    launch_gemm(xbuf, P.ff1_w, P.ff1_b, ffm, rows, DM, DFF, ACT_RELU, stream);
    launch_gemm(ffm, P.ff2_w, P.ff2_b, proj, rows, DFF, DM, ACT_NONE, stream);
    add_ln<<<(rows + 7) / 8, 256, 0, stream>>>(xbuf, proj, P.ln2_g, P.ln2_b, xbuf, rows);
  };

  enc_layer(hB, NN, LH, enc[0]);
  enc_layer(hB, NN, LH, enc[1]);

  // ---- context + queries ----
  ctx_mean<<<(BB * DM + 255) / 256, 256, 0, stream>>>(hB, ctx, LH);
  q_init<<<(M2 * DM + 255) / 256, 256, 0, stream>>>(queries, ctx, qb);

  enc_layer(qb, M2, TOUT, enc[0]);
  enc_layer(qb, M2, TOUT, enc[1]);

  // ---- decoder ----
  launch_gemm(qb, dec1_w, dec1_b, v1, M2, DM, 512, ACT_RELU, stream);
  {
    int blocks = ((NOUT / 16) + 7) / 8;          // 12288 strip-tiles, 8 waves/block
    gemm_dec2_wmma<<<blocks, 256, 0, stream>>>(v1, dec2_w, dec2_b, out);
  }
}